// SemiPool2d_4252017623210
// MI455X (gfx1250) — compile-verified
//
#include <hip/hip_runtime.h>

// ---------------------------------------------------------------------------
// SemiPool2d (max-plus semifield), valid padding, stride 2, kernel 7x7.
//   x:(8,64,224,224) f32, w:(1,64,7,7) f32  ->  out:(8,64,109,109) f32
// Memory-bound (~127MB traffic -> ~5.5us @ 23.3TB/s). Strategy:
//   * stage a nrowsx224 f32 stripe per block into LDS with ONE Tensor Data
//     Mover descriptor (TENSOR_LOAD_TO_LDS + s_wait_tensorcnt), falling back
//     to per-lane GLOBAL_LOAD_ASYNC_TO_LDS_B128 (ASYNCcnt), then sync copies;
//   * weights via uniform loads -> SGPRs (s_load);
//   * compute with v_add_f32 + v_max3_f32 trees; each lane owns a 2x2 output
//     patch (2 cols x 2 consecutive rows) so 9 staged rows feed 4 outputs and
//     every ds_load_b128 is bank-conflict-free (unit stride 1 per lane).
// WMMA is NOT applicable: max-plus is a tropical semiring, not (+,x).
// ---------------------------------------------------------------------------

#define AS1 __attribute__((address_space(1)))
#define AS3 __attribute__((address_space(3)))

typedef int          v4i __attribute__((ext_vector_type(4)));
typedef int          v8i __attribute__((ext_vector_type(8)));
typedef unsigned int v4u __attribute__((ext_vector_type(4)));

#if defined(__has_builtin)
#if __has_builtin(__builtin_amdgcn_tensor_load_to_lds) && \
    __has_builtin(__builtin_amdgcn_s_wait_tensorcnt)
#define USE_TDM 1
#endif
#if __has_builtin(__builtin_amdgcn_global_load_async_to_lds_b128) && \
    __has_builtin(__builtin_amdgcn_s_wait_asynccnt)
#define USE_ASYNC_LDS 1
#endif
#endif
#ifndef USE_TDM
#define USE_TDM 0
#endif
#ifndef USE_ASYNC_LDS
#define USE_ASYNC_LDS 0
#endif

namespace {
constexpr int BATCH = 8;
constexpr int C     = 64;
constexpr int H     = 224;
constexpr int W     = 224;
constexpr int K     = 7;
constexpr int S     = 2;
constexpr int HO    = (H - K) / S + 1;   // 109
constexpr int WO    = (W - K) / S + 1;   // 109
constexpr int TH    = 8;                 // output rows per block
constexpr int LROWS = TH * S + (K - S);  // 21 staged input rows
constexpr int LDSW  = W;                 // LDS row stride == tensor row (TDM linear)
constexpr int NV4   = W / 4;             // 56 float4 per input row
constexpr int NLOAD = LROWS * NV4;       // 1176 float4 tiles to stage (fallback)
}

__device__ __forceinline__ float max3f(float a, float b, float c) {
  return fmaxf(fmaxf(a, b), c);          // -> v_max3_f32 on AMDGPU
}

__global__ __launch_bounds__(256, 2)
void semipool2d_maxplus(const float* __restrict__ x,
                        const float* __restrict__ w,
                        float* __restrict__ out)
{
  __shared__ float smem[LROWS * LDSW + 16];   // +slack for b128 overshoot reads

  const int plane = blockIdx.y;          // b*C + c
  const int c     = plane & (C - 1);
  const int p0    = blockIdx.x * TH;     // first output row of this block
  const int r0    = p0 * S;              // first input row staged

  const float* __restrict__ xp = x + (size_t)plane * (H * W);
  const float* __restrict__ wc = w + c * (K * K);
  float*       __restrict__ op = out + (size_t)plane * (HO * WO);

  const int tid = threadIdx.x;
  int nrows = H - r0; if (nrows > LROWS) nrows = LROWS;   // 21, 16 for last stripe

#if USE_TDM
  // ---- one TDM descriptor stages the whole 224 x nrows f32 stripe ----
  if (tid < 32) {                         // wave 0 only (uniform branch)
    const unsigned long long gaddr =
        (unsigned long long)(size_t)(xp + (size_t)r0 * W);
    const unsigned laddr = (unsigned)(size_t)(AS3 float*)&smem[0];
    // D# group0: count=1 | lds_addr | global_addr[56:0] | type=2
    v4u g0 = { 1u, laddr, (unsigned)gaddr,
               (unsigned)(gaddr >> 32) | 0x80000000u };
    // D# group1: data_size=4B; tensor_dim0=224; tensor_dim1=224;
    //            tile_dim0=224; tile_dim1=nrows; tensor_dim0_stride=224
    v8i g1 = { (int)(2u << 16),          // workgroup_mask=0, data_size=2 (4B)
               (int)(224u << 16),        // tensor_dim0[15:0] in [63:48]
               (int)(224u << 16),        // tensor_dim1[15:0] in [95:80]
               (int)(224u << 16),        // tile_dim0 in [127:112]
               nrows,                    // tile_dim1 in [143:128]
               224,                      // tensor_dim0_stride[31:0]
               0, 0 };
    v4i z4 = { 0, 0, 0, 0 };             // groups 2/3 unused (2-D tile)
#if defined(__clang_major__) && __clang_major__ >= 23
    v8i z8 = { 0, 0, 0, 0, 0, 0, 0, 0 };
    __builtin_amdgcn_tensor_load_to_lds(g0, g1, z4, z4, z8, 0);  // clang-23 form
#else
    __builtin_amdgcn_tensor_load_to_lds(g0, g1, z4, z4, 0);      // ROCm 7.2 form
#endif
    __builtin_amdgcn_s_wait_tensorcnt(0);
  }
  __syncthreads();
#else
  // ---- fallback: per-lane staging (async-to-LDS if available) ----
  for (int idx = tid; idx < NLOAD; idx += 256) {
    const int r  = idx / NV4;
    const int c4 = idx - r * NV4;
    int gr = r0 + r;                     // clamp: rows past the image are unused
    if (gr > H - 1) gr = H - 1;
    const float* gp = xp + gr * W + c4 * 4;
    float*       lp = &smem[r * LDSW + c4 * 4];
#if USE_ASYNC_LDS
    __builtin_amdgcn_global_load_async_to_lds_b128(
        (AS1 v4i*)gp, (AS3 v4i*)lp, 0, 0);
#else
    *(float4*)lp = *(const float4*)gp;
#endif
  }
#if USE_ASYNC_LDS
  __builtin_amdgcn_s_wait_asynccnt(0);
#endif
  __syncthreads();
#endif

  // ---- uniform weight loads: lower to s_load (SGPR-resident) ----
  float wv[K * K];
#pragma unroll
  for (int k = 0; k < K * K; ++k) wv[k] = wc[k];

  // ---- compute: each lane owns a 2x2 output patch ----
  // wave pair (half) splits the 109 output cols: half 0 -> cols 0..63,
  // half 1 -> cols 64..108 (lanes 0..22). ty0 picks 2 consecutive out rows.
  const int lane  = tid & 31;
  const int wave  = tid >> 5;
  const int half  = wave & 1;
  const int ty0   = wave >> 1;                 // 0..3
  const int q0    = half * 64 + lane * 2;      // output col base (2 cols/lane)
  const int pa    = p0 + 2 * ty0;              // first of 2 output rows
  const int rbase = 4 * ty0;                   // first staged row this lane reads

  if (pa < HO && q0 < WO) {
    float aA0 = -1e30f, aA1 = -1e30f, aB0 = -1e30f, aB1 = -1e30f;

#pragma unroll
    for (int i2 = 0; i2 < K + S; ++i2) {       // 9 staged rows -> 2 output rows
      const float* lr = &smem[(rbase + i2) * LDSW + q0 * S];
      float v[12];                             // dwords 2*q0 .. 2*q0+11
      *(float4*)&v[0] = *(const float4*)(lr);
      *(float4*)&v[4] = *(const float4*)(lr + 4);
      *(float4*)&v[8] = *(const float4*)(lr + 8);

      if (i2 < K) {                            // feeds output row pa (w-row i2)
        const float* wr = &wv[i2 * K];
        float m0 = max3f(v[0] + wr[0], v[1] + wr[1], v[2] + wr[2]);
        m0 = max3f(m0, v[3] + wr[3], v[4] + wr[4]);
        m0 = max3f(m0, v[5] + wr[5], v[6] + wr[6]);
        aA0 = fmaxf(aA0, m0);
        float m1 = max3f(v[2] + wr[0], v[3] + wr[1], v[4] + wr[2]);
        m1 = max3f(m1, v[5] + wr[3], v[6] + wr[4]);
        m1 = max3f(m1, v[7] + wr[5], v[8] + wr[6]);
        aA1 = fmaxf(aA1, m1);
      }
      if (i2 >= S) {                           // feeds output row pa+1 (w-row i2-2)
        const float* wr = &wv[(i2 - S) * K];
        float m0 = max3f(v[0] + wr[0], v[1] + wr[1], v[2] + wr[2]);
        m0 = max3f(m0, v[3] + wr[3], v[4] + wr[4]);
        m0 = max3f(m0, v[5] + wr[5], v[6] + wr[6]);
        aB0 = fmaxf(aB0, m0);
        float m1 = max3f(v[2] + wr[0], v[3] + wr[1], v[4] + wr[2]);
        m1 = max3f(m1, v[5] + wr[3], v[6] + wr[4]);
        m1 = max3f(m1, v[7] + wr[5], v[8] + wr[6]);
        aB1 = fmaxf(aB1, m1);
      }
    }

    float* rowA = op + pa * WO;
    rowA[q0] = aA0;
    if (q0 + 1 < WO) rowA[q0 + 1] = aA1;
    const int pb = pa + 1;
    if (pb < HO) {
      float* rowB = op + pb * WO;
      rowB[q0] = aB0;
      if (q0 + 1 < WO) rowB[q0 + 1] = aB1;
    }
  }
}

extern "C" void kernel_launch(void* const* d_in, const int* in_sizes, int n_in,
                              void* d_out, int out_size, void* d_ws, size_t ws_size,
                              hipStream_t stream) {
  (void)in_sizes; (void)n_in; (void)out_size; (void)d_ws; (void)ws_size;
  const float* x = (const float*)d_in[0];
  const float* w = (const float*)d_in[1];
  float* out     = (float*)d_out;

  dim3 grid((HO + TH - 1) / TH, BATCH * C);   // (14, 512)
  semipool2d_maxplus<<<grid, 256, 0, stream>>>(x, w, out);
}